// HeadAttention_50732153700532
// MI455X (gfx1250) — compile-verified
//
#include <hip/hip_runtime.h>
#include <math.h>

// ---------------------------------------------------------------------------
// Causal single-head attention for MI455X (gfx1250, wave32).
//   y = softmax(mask(QK^T / sqrt(H))) V,  Q/K/V = x*W + b
// B=8 S=2048 E=1024 H=64, fp32 end-to-end.
//
// - flash-attention tiling: ~72MB total HBM traffic (~3us @ 23.3 TB/s),
//   never materializes the S x S score matrix.
// - exact-fp32 V_WMMA_F32_16X16X4_F32 for all matmuls.
// - LDS tiles double-buffered and filled with gfx1250 async global->LDS
//   copies (global_load_async_to_lds_b128, ASYNCcnt pipelining).
// - W pre-transposed once so every LDS fragment read is a contiguous b64.
// ---------------------------------------------------------------------------

typedef __attribute__((ext_vector_type(2))) float v2f;
typedef __attribute__((ext_vector_type(4))) float v4f;
typedef __attribute__((ext_vector_type(8))) float v8f;

#define DEVINL __device__ __forceinline__

constexpr int Bn = 8, Sn = 2048, En = 1024, Hn = 64;
#define ATT_SCALE 0.125f  // 1/sqrt(64)

// ---- gfx1250 async global->LDS copy (inline asm) -------
DEVINL void g2l_16B(const float* g, float* l) {
  unsigned lds_off = (unsigned)(unsigned long long)
      (__attribute__((address_space(3))) float*)l;
  unsigned long long gaddr = (unsigned long long)g;
  asm volatile("global_load_async_to_lds_b128 %0, %1, off"
               :: "v"(lds_off), "v"(gaddr) : "memory");
}

DEVINL void async_wait0() {
  asm volatile("s_wait_asynccnt 0x0" ::: "memory");
}

DEVINL void lds_order() {
#if __has_builtin(__builtin_amdgcn_s_wait_dscnt)
  __builtin_amdgcn_s_wait_dscnt(0);
#else
  asm volatile("" ::: "memory");  // same-wave LDS ops are in-order in HW
#endif
}

// ---- exact-fp32 WMMA: D = A(16x4) * B(4x16) + C ---------------------------
DEVINL v8f wmma4(v2f a, v2f b, v8f c) {
  return __builtin_amdgcn_wmma_f32_16x16x4_f32(
      /*neg_a=*/false, a, /*neg_b=*/false, b,
      /*c_mod=*/(short)0, c, /*reuse_a=*/false, /*reuse_b=*/false);
}

// ===========================================================================
// Kernel 0: one-time weight transpose  w[E][H] -> wt[H][E]  (3 matrices).
// 768 KB total; negligible vs the 67 MB x stream, buys contiguous-b64 LDS
// fragment reads in the projection hot loop.
// ===========================================================================
__global__ __launch_bounds__(256) void wtrans_kernel(
    const float* __restrict__ wq, const float* __restrict__ wk,
    const float* __restrict__ wv, float* __restrict__ wt) {
  const float* w = (blockIdx.y == 0) ? wq : (blockIdx.y == 1) ? wk : wv;
  float* o = wt + (size_t)blockIdx.y * Hn * En;
  const int idx = blockIdx.x * 256 + threadIdx.x;  // over E*H = 65536
  const int e = idx >> 6, h = idx & 63;            // coalesced read over h
  o[(size_t)h * En + e] = w[(size_t)e * Hn + h];
}

// ===========================================================================
// Kernel 1: fused Q/K/V projection.  grid = (M/32, 3), block = 128 (4 waves).
// Each block: one 32-row M tile; wave w owns N tile w (H=64 = 4 tiles) and
// two 16-row sub-tiles sharing every B fragment (32 WMMA / ~16 DS per chunk).
// K streamed 64-wide through double-buffered LDS via async copies.
// ===========================================================================
__global__ __launch_bounds__(128) void qkv_proj_kernel(
    const float* __restrict__ x, const float* __restrict__ wt,  // wt:[3][H][E]
    const float* __restrict__ bq, const float* __restrict__ bk,
    const float* __restrict__ bv,
    float* __restrict__ qo, float* __restrict__ ko, float* __restrict__ vo) {
  __shared__ float sX[2][32 * 64];   // [m][k]   2 x 8 KB
  __shared__ float sW[2][64 * 64];   // [n][k]   2 x 16 KB

  const float* bias; float* out;
  if (blockIdx.y == 0)      { bias = bq; out = qo; }
  else if (blockIdx.y == 1) { bias = bk; out = ko; }
  else                      { bias = bv; out = vo; }
  const float* wtm = wt + (size_t)blockIdx.y * Hn * En;  // [H][E]

  const int tid  = threadIdx.x;
  const int wid  = tid >> 5;      // wave id = N tile
  const int lane = tid & 31;
  const int hs   = lane >> 4;     // lane half (K split in A/B fragments)
  const int l16  = lane & 15;
  const int mbase = blockIdx.x * 32;

  auto load_tiles = [&](int kc, int buf) {
    // X tile 32x64 floats: 512 float4, 4 per thread (async -> LDS)
    for (int i = tid; i < 512; i += 128) {
      const int m = i >> 4, f4 = i & 15;
      g2l_16B(&x[(size_t)(mbase + m) * En + kc + f4 * 4],
              &sX[buf][m * 64 + f4 * 4]);
    }
    // Wt tile 64(n) x 64(k): 1024 float4, 8 per thread
    for (int i = tid; i < 1024; i += 128) {
      const int n = i >> 4, f4 = i & 15;
      g2l_16B(&wtm[(size_t)n * En + kc + f4 * 4],
              &sW[buf][n * 64 + f4 * 4]);
    }
  };

  v8f acc0 = {}, acc1 = {};

  load_tiles(0, 0);                        // prologue prefetch
  constexpr int NCHUNK = En / 64;          // 16
  for (int ch = 0; ch < NCHUNK; ++ch) {
    const int buf = ch & 1;
    async_wait0();      // this wave's prefetch of `buf` complete
    __syncthreads();    // all waves loaded `buf`; all done reading buf^1
    if (ch + 1 < NCHUNK) load_tiles((ch + 1) * 64, buf ^ 1);  // overlap DMA

#pragma unroll
    for (int c = 0; c < 16; ++c) {
      const int k0 = c * 4 + hs * 2;
      // B 4x16 (lane=col n, K pair {k0,k0+1}) from transposed W: one b64
      v2f bf = *(const v2f*)&sW[buf][(wid * 16 + l16) * 64 + k0];
      // A 16x4 fragments for both M sub-tiles
      v2f a0 = *(const v2f*)&sX[buf][l16 * 64 + k0];
      v2f a1 = *(const v2f*)&sX[buf][(16 + l16) * 64 + k0];
      acc0 = wmma4(a0, bf, acc0);
      acc1 = wmma4(a1, bf, acc1);
    }
  }

  // Epilogue: C layout (VGPR r -> row r + 8*hs, col = l16), fuse bias.
  const float bb = bias[wid * 16 + l16];
#pragma unroll
  for (int r = 0; r < 8; ++r) {
    const int row0 = mbase + r + hs * 8;
    const int row1 = row0 + 16;
    out[(size_t)row0 * Hn + wid * 16 + l16] = acc0[r] + bb;
    out[(size_t)row1 * Hn + wid * 16 + l16] = acc1[r] + bb;
  }
}

// ===========================================================================
// Kernel 2: flash attention.  grid = (S/64, B), block = 128 (4 waves).
// Wave w owns Q tile qt = blockIdx.x*4 + w (16 query rows, Q in registers).
// Block streams 16-key K/V tiles through double-buffered LDS (async DMA
// overlapped with compute); per tile: 16 WMMA scores, causal mask + online
// softmax, 16 WMMA PV.
// ===========================================================================
__global__ __launch_bounds__(128) void flash_attn_kernel(
    const float* __restrict__ qm, const float* __restrict__ km,
    const float* __restrict__ vm, float* __restrict__ out) {
  __shared__ float sK[2][16 * 64];    // [key][h]  2 x 4 KB
  __shared__ float sV[2][16 * 64];    // [key][h]  2 x 4 KB
  __shared__ float sP[4][16 * 16];    // per-wave probs staging, 4 KB

  const int tid  = threadIdx.x;
  const int wid  = tid >> 5;
  const int lane = tid & 31;
  const int hs   = lane >> 4;
  const int l16  = lane & 15;

  const int bidx  = blockIdx.y;
  const int qt    = blockIdx.x * 4 + wid;
  const int qbase = qt * 16;
  const size_t boff = (size_t)bidx * Sn * Hn;

  auto load_kv = [&](int kt, int buf) {
    for (int i = tid; i < 256; i += 128) {
      const int kr = i >> 4, f4 = i & 15;
      const size_t g = boff + (size_t)(kt * 16 + kr) * Hn + f4 * 4;
      g2l_16B(&km[g], &sK[buf][kr * 64 + f4 * 4]);
      g2l_16B(&vm[g], &sV[buf][kr * 64 + f4 * 4]);
    }
  };

  // Q tile 16x64 as 16 A-fragments (float2/lane), kept in VGPRs.
  v2f qa[16];
#pragma unroll
  for (int c = 0; c < 16; ++c)
    qa[c] = *(const v2f*)&qm[boff + (size_t)(qbase + l16) * Hn + c * 4 + hs * 2];

  v8f o[4] = {};                 // output accum: 4 N-tiles over H
  float m_r[8], l_r[8];          // online-softmax stats, rows r + 8*hs
#pragma unroll
  for (int r = 0; r < 8; ++r) { m_r[r] = -__builtin_inff(); l_r[r] = 0.f; }

  const int ktmax = blockIdx.x * 4 + 3;  // uniform over block (barrier-legal)
  load_kv(0, 0);                         // prologue prefetch
  for (int kt = 0; kt <= ktmax; ++kt) {
    const int buf = kt & 1;
    async_wait0();     // K/V tile `buf` resident
    __syncthreads();   // all waves see it; all done reading buf^1
    if (kt < ktmax) load_kv(kt + 1, buf ^ 1);  // overlap next tile's DMA

    if (kt <= qt) {  // wave-uniform: EXEC stays all-ones around WMMA
      // ---- S = Q * K^T : B[kk=h][n=key] = K[key][h] -> contiguous float2
      v8f s = {};
#pragma unroll
      for (int c = 0; c < 16; ++c) {
        const int k0 = c * 4 + hs * 2;
        v2f bf = *(const v2f*)&sK[buf][l16 * 64 + k0];
        s = wmma4(qa[c], bf, s);
      }

      // ---- scale + causal mask (C layout: row = r+8*hs, col lane = key)
      const int col = kt * 16 + l16;
#pragma unroll
      for (int r = 0; r < 8; ++r) {
        const int row = qbase + r + hs * 8;
        s[r] = (col > row) ? -__builtin_inff() : s[r] * ATT_SCALE;
      }

      // ---- online softmax: reductions over the 16 lanes of each C half
#pragma unroll
      for (int r = 0; r < 8; ++r) {
        float mx = s[r];
#pragma unroll
        for (int d = 1; d < 16; d <<= 1) mx = fmaxf(mx, __shfl_xor(mx, d, 32));
        const float mnew  = fmaxf(m_r[r], mx);
        const float alpha = __expf(m_r[r] - mnew);
        const float p     = __expf(s[r] - mnew);
        float ps = p;
#pragma unroll
        for (int d = 1; d < 16; d <<= 1) ps += __shfl_xor(ps, d, 32);
        l_r[r] = l_r[r] * alpha + ps;
        m_r[r] = mnew;
#pragma unroll
        for (int t = 0; t < 4; ++t) o[t][r] *= alpha;
        s[r] = p;
      }

      // ---- restripe P: C layout -> row-major LDS -> A fragments
#pragma unroll
      for (int r = 0; r < 8; ++r)
        sP[wid][(r + hs * 8) * 16 + l16] = s[r];
      lds_order();

      // ---- O += P * V : 4 key-chunks x 4 N-tiles = 16 WMMA
#pragma unroll
      for (int c = 0; c < 4; ++c) {
        const int k0 = c * 4 + hs * 2;
        v2f pa = *(const v2f*)&sP[wid][l16 * 16 + k0];
#pragma unroll
        for (int t = 0; t < 4; ++t) {
          v2f bf;
          bf.x = sV[buf][(k0 + 0) * 64 + t * 16 + l16];
          bf.y = sV[buf][(k0 + 1) * 64 + t * 16 + l16];
          o[t] = wmma4(pa, bf, o[t]);
        }
      }
    }
    // no trailing barrier: top-of-loop wait+barrier protects buf reuse
  }

  // ---- epilogue: y = O / l
#pragma unroll
  for (int r = 0; r < 8; ++r) {
    const float inv = 1.0f / l_r[r];
    const int row = qbase + r + hs * 8;
#pragma unroll
    for (int t = 0; t < 4; ++t)
      out[boff + (size_t)row * Hn + t * 16 + l16] = o[t][r] * inv;
  }
}

// ===========================================================================
extern "C" void kernel_launch(void* const* d_in, const int* in_sizes, int n_in,
                              void* d_out, int out_size, void* d_ws,
                              size_t ws_size, hipStream_t stream) {
  const float* x  = (const float*)d_in[0];
  const float* wq = (const float*)d_in[1];
  const float* bq = (const float*)d_in[2];
  const float* wk = (const float*)d_in[3];
  const float* bk = (const float*)d_in[4];
  const float* wv = (const float*)d_in[5];
  const float* bv = (const float*)d_in[6];

  float* qbuf  = (float*)d_ws;                       // 4 MB
  float* kbuf  = qbuf + (size_t)Bn * Sn * Hn;        // 4 MB
  float* vbuf  = kbuf + (size_t)Bn * Sn * Hn;        // 4 MB
  float* wtbuf = vbuf + (size_t)Bn * Sn * Hn;        // 3 x 256 KB

  dim3 g0((En * Hn) / 256, 3), b0(256);
  wtrans_kernel<<<g0, b0, 0, stream>>>(wq, wk, wv, wtbuf);

  dim3 g1((Bn * Sn) / 32, 3), b1(128);
  qkv_proj_kernel<<<g1, b1, 0, stream>>>(x, wtbuf, bq, bk, bv,
                                         qbuf, kbuf, vbuf);

  dim3 g2(Sn / 64, Bn), b2(128);
  flash_attn_kernel<<<g2, b2, 0, stream>>>(qbuf, kbuf, vbuf, (float*)d_out);
}